// SyntaxGCN_73718818669021
// MI455X (gfx1250) — compile-verified
//
#include <hip/hip_runtime.h>
#include <hip/hip_bf16.h>

typedef __attribute__((ext_vector_type(2))) float v2f;
typedef __attribute__((ext_vector_type(8))) float v8f;

#define HID 64
#define IN_DIM 256
#define NUM_GRAPHS 512

// ---------------------------------------------------------------- utilities
__global__ void k_zero(float* p, int n) {
    int i = blockIdx.x * blockDim.x + threadIdx.x;
    if (i < n) p[i] = 0.0f;
}

// ------------------------------------------------------------ degree pass
__global__ void k_edge_deg(const int* __restrict__ ei, int E, float* __restrict__ deg) {
    int e = blockIdx.x * blockDim.x + threadIdx.x;
    if (e >= E) return;
    int d = ei[E + e];            // dst row of edge_index[2,E]
    unsafeAtomicAdd(&deg[d], 1.0f);
}

__global__ void k_finalize_deg(const float* __restrict__ deg,
                               float* __restrict__ inv_sqrt,
                               float* __restrict__ inv_deg, int N) {
    int n = blockIdx.x * blockDim.x + threadIdx.x;
    if (n >= N) return;
    float d = deg[n] + 1.0f;      // + self loop
    inv_sqrt[n] = __frsqrt_rn(d);
    inv_deg[n]  = 1.0f / d;
}

// ------------------------------------------------------------ dense GEMM
// out[M x 64] = A[M x K] * W[K x 64], f32, V_WMMA_F32_16X16X4_F32.
// Block = 128 threads = 4 waves; wave w computes column tile w (16 cols),
// blockIdx.x selects the 16-row M tile.
__global__ void k_gemm_wmma(const float* __restrict__ A,
                            const float* __restrict__ W,
                            float* __restrict__ out, int M, int K) {
    const int lane = threadIdx.x & 31;
    const int wave = threadIdx.x >> 5;          // N tile: 0..3
    const int mt   = blockIdx.x;                // M tile
    const int r    = lane & 15;
    const int hi   = lane >> 4;                 // 0: K={0,1}  1: K={2,3}
    const int kk   = hi * 2;

    int row = mt * 16 + r;
    int rowc = row < M ? row : (M - 1);         // clamp (EXEC must stay all-1s)
    const float* Arow = A + (size_t)rowc * K;
    const int col = wave * 16 + r;

    v8f c = {};
    for (int k0 = 0; k0 < K; k0 += 4) {
        v2f a, b;
        // A fragment: lane holds A[row, k0+kk], A[row, k0+kk+1] (contiguous)
        a.x = Arow[k0 + kk];
        a.y = Arow[k0 + kk + 1];
        // B fragment: lane holds W[k0+kk, col], W[k0+kk+1, col]
        b.x = W[(size_t)(k0 + kk) * HID + col];
        b.y = W[(size_t)(k0 + kk + 1) * HID + col];
        c = __builtin_amdgcn_wmma_f32_16x16x4_f32(
                /*neg_a=*/false, a, /*neg_b=*/false, b,
                /*c_mod=*/(short)0, c, /*reuse_a=*/false, /*reuse_b=*/false);
    }

    // D layout: VGPR i -> M = i (lanes 0-15) / i+8 (lanes 16-31), N = lane&15
    #pragma unroll
    for (int i = 0; i < 8; ++i) {
        int orow = mt * 16 + i + 8 * hi;
        if (orow < M)
            out[(size_t)orow * HID + wave * 16 + r] = c[i];
    }
}

// --------------------------------------------- self-loop term + bias init
// agg[n,j] = h[n,j] * inv_deg[n] + b[j]
__global__ void k_init_agg(const float* __restrict__ h,
                           const float* __restrict__ inv_deg,
                           const float* __restrict__ bias,
                           float* __restrict__ agg, int N) {
    int idx = blockIdx.x * blockDim.x + threadIdx.x;
    if (idx >= N * HID) return;
    int n = idx >> 6, j = idx & 63;
    agg[idx] = h[idx] * inv_deg[n] + bias[j];
}

// --------------------------------------------------- edge scatter (1 wave/edge)
__global__ void k_edge_agg(const int* __restrict__ ei, int E,
                           const float* __restrict__ inv_sqrt,
                           const float* __restrict__ h,
                           float* __restrict__ agg) {
    int e = blockIdx.x * (blockDim.x >> 5) + (threadIdx.x >> 5);
    if (e >= E) return;
    int lane = threadIdx.x & 31;
    int s = ei[e];
    int d = ei[E + e];
    float nrm = inv_sqrt[s] * inv_sqrt[d];
    const float2 hv = *(const float2*)&h[(size_t)s * HID + 2 * lane];
    float* dp = &agg[(size_t)d * HID + 2 * lane];
    unsafeAtomicAdd(dp,     hv.x * nrm);
    unsafeAtomicAdd(dp + 1, hv.y * nrm);
}

__global__ void k_relu(float* p, int n) {
    int i = blockIdx.x * blockDim.x + threadIdx.x;
    if (i < n) p[i] = fmaxf(p[i], 0.0f);
}

// ------------------------------------------------------ pooling (1 wave/node)
__global__ void k_pool(const float* __restrict__ h, const int* __restrict__ batch,
                       float* __restrict__ pooled, float* __restrict__ counts, int N) {
    int n = blockIdx.x * (blockDim.x >> 5) + (threadIdx.x >> 5);
    if (n >= N) return;
    int lane = threadIdx.x & 31;
    int g = batch[n];
    const float2 hv = *(const float2*)&h[(size_t)n * HID + 2 * lane];
    float* dp = &pooled[(size_t)g * HID + 2 * lane];
    unsafeAtomicAdd(dp,     hv.x);
    unsafeAtomicAdd(dp + 1, hv.y);
    if (lane == 0) unsafeAtomicAdd(&counts[g], 1.0f);
}

// ----------------------------------------------- final linear (1 wave/graph)
__global__ void k_final(const float* __restrict__ pooled,
                        const float* __restrict__ counts,
                        const float* __restrict__ Wlin,
                        const float* __restrict__ blin,
                        float* __restrict__ out, int G) {
    int g = blockIdx.x * (blockDim.x >> 5) + (threadIdx.x >> 5);
    if (g >= G) return;
    int lane = threadIdx.x & 31;
    float inv = 1.0f / fmaxf(counts[g], 1.0f);
    float v = pooled[(size_t)g * HID + lane]      * Wlin[lane]
            + pooled[(size_t)g * HID + lane + 32] * Wlin[lane + 32];
    v *= inv;
    #pragma unroll
    for (int off = 16; off > 0; off >>= 1) v += __shfl_xor(v, off, 32);
    if (lane == 0) out[g] = v + blin[0];
}

// ================================================================== launch
extern "C" void kernel_launch(void* const* d_in, const int* in_sizes, int n_in,
                              void* d_out, int out_size, void* d_ws, size_t ws_size,
                              hipStream_t stream) {
    const float* x    = (const float*)d_in[0];
    const int*   ei   = (const int*)  d_in[1];
    const int*   batch= (const int*)  d_in[2];
    const float* W1   = (const float*)d_in[3];
    const float* b1   = (const float*)d_in[4];
    const float* W2   = (const float*)d_in[5];
    const float* b2   = (const float*)d_in[6];
    const float* Wlin = (const float*)d_in[7];
    const float* blin = (const float*)d_in[8];
    float* out = (float*)d_out;

    const int N = in_sizes[0] / IN_DIM;
    const int E = in_sizes[1] / 2;
    const int G = NUM_GRAPHS;

    // workspace layout (floats)
    float* ws       = (float*)d_ws;
    float* deg      = ws;                       // N
    float* inv_sqrt = ws + (size_t)N;           // N
    float* inv_deg  = ws + (size_t)2 * N;       // N
    float* bufA     = ws + (size_t)3 * N;       // N*64  (h = X@W)
    float* bufB     = bufA + (size_t)N * HID;   // N*64  (agg / layer output)
    float* pooled   = bufB + (size_t)N * HID;   // G*64
    float* counts   = pooled + (size_t)G * HID; // G

    const int NT = 256;
    const int mtiles = (N + 15) / 16;
    const int wavesPB = NT / 32;

    // --- degree ---
    k_zero<<<(N + NT - 1) / NT, NT, 0, stream>>>(deg, N);
    k_zero<<<(G * HID + G + NT - 1) / NT, NT, 0, stream>>>(pooled, G * HID + G);
    k_edge_deg<<<(E + NT - 1) / NT, NT, 0, stream>>>(ei, E, deg);
    k_finalize_deg<<<(N + NT - 1) / NT, NT, 0, stream>>>(deg, inv_sqrt, inv_deg, N);

    // --- layer 1 ---
    k_gemm_wmma<<<mtiles, 128, 0, stream>>>(x, W1, bufA, N, IN_DIM);
    k_init_agg<<<(N * HID + NT - 1) / NT, NT, 0, stream>>>(bufA, inv_deg, b1, bufB, N);
    k_edge_agg<<<(E + wavesPB - 1) / wavesPB, NT, 0, stream>>>(ei, E, inv_sqrt, bufA, bufB);
    k_relu<<<(N * HID + NT - 1) / NT, NT, 0, stream>>>(bufB, N * HID);

    // --- layer 2 (bufB = h1, bufA = h1@W2, then bufB reused as agg) ---
    k_gemm_wmma<<<mtiles, 128, 0, stream>>>(bufB, W2, bufA, N, HID);
    k_init_agg<<<(N * HID + NT - 1) / NT, NT, 0, stream>>>(bufA, inv_deg, b2, bufB, N);
    k_edge_agg<<<(E + wavesPB - 1) / wavesPB, NT, 0, stream>>>(ei, E, inv_sqrt, bufA, bufB);
    k_relu<<<(N * HID + NT - 1) / NT, NT, 0, stream>>>(bufB, N * HID);

    // --- pooling + final linear ---
    k_pool<<<(N + wavesPB - 1) / wavesPB, NT, 0, stream>>>(bufB, batch, pooled, counts, N);
    k_final<<<(G + wavesPB - 1) / wavesPB, NT, 0, stream>>>(pooled, counts, Wlin, blin, out, G);
}